// UnifiedFlowFrag_7421703488097
// MI455X (gfx1250) — compile-verified
//
#include <hip/hip_runtime.h>
#include <hip/hip_bf16.h>
#include <stdint.h>

#define GLOBAL_AS __attribute__((address_space(1)))
#define LDS_AS    __attribute__((address_space(3)))

// Matches the builtin's parameter pointee type exactly:
//   'int __attribute__((__vector_size__(4 * sizeof(int))))'
typedef int ai4 __attribute__((vector_size(4 * sizeof(int))));

// CDNA5 async global->LDS path (async data-mover class ops).
// Guarded: falls back to plain b128 loads if the builtin is absent.
#if defined(__HIP_DEVICE_COMPILE__) && __has_builtin(__builtin_amdgcn_global_load_async_to_lds_b128)
#define USE_ASYNC_LDS 1
#else
#define USE_ASYNC_LDS 0
#endif

// ---------------------------------------------------------------------------
// Accumulator layout per fragment (16 floats = 64B, padded for alignment):
//   [0..2]  sum f            [3..5]  sum (r x f)
//   [6..11] sum xx,yy,zz,xy,xz,yz    [12] count     [13..15] pad
// ---------------------------------------------------------------------------
#define ACC_STRIDE 16

// First kernel in the file -> its prologue appears in the disasm snippet.
__global__ __launch_bounds__(256) void frag_accumulate(
    const float* __restrict__ f_atom, const float* __restrict__ atom_pos,
    const float* __restrict__ T_frag, const int* __restrict__ frag_id,
    float* __restrict__ acc, int n_atom) {
  __shared__ alignas(16) float sh_f[256 * 3];
  __shared__ alignas(16) float sh_p[256 * 3];

  const int lt   = threadIdx.x;
  const int base = blockIdx.x * 256;
  const int gid  = base + lt;

  if (base + 256 <= n_atom) {            // full tile (uniform per block)
    const size_t fb = (size_t)base * 3;  // 3072B tile per stream, 16B aligned
#if USE_ASYNC_LDS
    if (lt < 192) {
      // 192 x b128 async transfers per stream; written by the async data
      // mover directly into LDS, tracked on ASYNCcnt.
      float* gf = const_cast<float*>(f_atom) + fb + (size_t)lt * 4;
      float* gp = const_cast<float*>(atom_pos) + fb + (size_t)lt * 4;
      __builtin_amdgcn_global_load_async_to_lds_b128(
          (GLOBAL_AS ai4*)gf, (LDS_AS ai4*)(sh_f + lt * 4), 0, 0);
      __builtin_amdgcn_global_load_async_to_lds_b128(
          (GLOBAL_AS ai4*)gp, (LDS_AS ai4*)(sh_p + lt * 4), 0, 0);
    }
#if __has_builtin(__builtin_amdgcn_s_wait_asynccnt)
    __builtin_amdgcn_s_wait_asynccnt(0);
#else
    asm volatile("s_wait_asynccnt 0x0" ::: "memory");
#endif
    __syncthreads();
#else
    if (lt < 192) {
      ((float4*)sh_f)[lt] = ((const float4*)(f_atom + fb))[lt];
      ((float4*)sh_p)[lt] = ((const float4*)(atom_pos + fb))[lt];
    }
    __syncthreads();
#endif
  } else {
    // Tail tile: stage guarded scalar loads into LDS so the consumption path
    // below is LDS-only in BOTH branches (prevents the compiler merging LDS
    // and global reads into generic flat loads that tie up DScnt+LOADcnt).
    if (gid < n_atom) {
      sh_f[lt * 3 + 0] = f_atom[(size_t)gid * 3 + 0];
      sh_f[lt * 3 + 1] = f_atom[(size_t)gid * 3 + 1];
      sh_f[lt * 3 + 2] = f_atom[(size_t)gid * 3 + 2];
      sh_p[lt * 3 + 0] = atom_pos[(size_t)gid * 3 + 0];
      sh_p[lt * 3 + 1] = atom_pos[(size_t)gid * 3 + 1];
      sh_p[lt * 3 + 2] = atom_pos[(size_t)gid * 3 + 2];
    }
    __syncthreads();
  }

  // Single read path: stride-3 b32 LDS reads, gcd(3,64)=1 -> conflict-free.
  const float fx = sh_f[lt * 3 + 0];
  const float fy = sh_f[lt * 3 + 1];
  const float fz = sh_f[lt * 3 + 2];
  const float px = sh_p[lt * 3 + 0];
  const float py = sh_p[lt * 3 + 1];
  const float pz = sh_p[lt * 3 + 2];

  if (gid >= n_atom) return;

  const int id = frag_id[gid];
  const float Tx = T_frag[(size_t)id * 3 + 0];
  const float Ty = T_frag[(size_t)id * 3 + 1];
  const float Tz = T_frag[(size_t)id * 3 + 2];
  const float rx = px - Tx, ry = py - Ty, rz = pz - Tz;
  const float cx = ry * fz - rz * fy;
  const float cy = rz * fx - rx * fz;
  const float cz = rx * fy - ry * fx;

  float* a = acc + (size_t)id * ACC_STRIDE;  // 12.8MB: L2-resident atomics
  atomicAdd(a + 0, fx);       atomicAdd(a + 1, fy);       atomicAdd(a + 2, fz);
  atomicAdd(a + 3, cx);       atomicAdd(a + 4, cy);       atomicAdd(a + 5, cz);
  atomicAdd(a + 6, rx * rx);  atomicAdd(a + 7, ry * ry);  atomicAdd(a + 8, rz * rz);
  atomicAdd(a + 9, rx * ry);  atomicAdd(a + 10, rx * rz); atomicAdd(a + 11, ry * rz);
  atomicAdd(a + 12, 1.0f);
}

__global__ __launch_bounds__(256) void ws_zero(float* __restrict__ p, int n) {
  int i = blockIdx.x * 256 + threadIdx.x;
  if (i < n) p[i] = 0.0f;
}

// ----------------------- 3x3 symmetric eigensolver -------------------------
__device__ __forceinline__ void cross3(const float* a, const float* b, float* o) {
  o[0] = a[1] * b[2] - a[2] * b[1];
  o[1] = a[2] * b[0] - a[0] * b[2];
  o[2] = a[0] * b[1] - a[1] * b[0];
}
__device__ __forceinline__ float dot3(const float* a, const float* b) {
  return a[0] * b[0] + a[1] * b[1] + a[2] * b[2];
}
// A packed: a00,a11,a22,a01,a02,a12
__device__ __forceinline__ void sym_eigvec(const float A[6], float lam, float v[3]) {
  float r0[3] = {A[0] - lam, A[3], A[4]};
  float r1[3] = {A[3], A[1] - lam, A[5]};
  float r2[3] = {A[4], A[5], A[2] - lam};
  float c0[3], c1[3], c2[3];
  cross3(r0, r1, c0); cross3(r0, r2, c1); cross3(r1, r2, c2);
  float n0 = dot3(c0, c0), n1 = dot3(c1, c1), n2 = dot3(c2, c2);
  const float* best = c0; float nb = n0;
  if (n1 > nb) { nb = n1; best = c1; }
  if (n2 > nb) { nb = n2; best = c2; }
  if (nb < 1e-30f) { v[0] = 1.0f; v[1] = 0.0f; v[2] = 0.0f; return; }
  float inv = rsqrtf(nb);
  v[0] = best[0] * inv; v[1] = best[1] * inv; v[2] = best[2] * inv;
}

__global__ __launch_bounds__(256) void frag_finalize(
    const float* __restrict__ acc, const int* __restrict__ frag_sizes,
    float* __restrict__ out, int n_frag) {
  const int f = blockIdx.x * 256 + threadIdx.x;
  if (f >= n_frag) return;
  const float* a = acc + (size_t)f * ACC_STRIDE;
  const float sfx = a[0], sfy = a[1], sfz = a[2];
  const float tx = a[3], ty = a[4], tz = a[5];
  const float xx = a[6], yy = a[7], zz = a[8];
  const float xy = a[9], xz = a[10], yz = a[11];
  const float cnt = a[12];

  float* v_out = out;
  float* w_out = out + (size_t)3 * n_frag;
  float* P_out = out + (size_t)6 * n_frag;

  const float inv_d = 1.0f / fmaxf(cnt, 1.0f);
  v_out[3 * f + 0] = sfx * inv_d;
  v_out[3 * f + 1] = sfy * inv_d;
  v_out[3 * f + 2] = sfz * inv_d;

  // Inertia tensor: I = (xx+yy+zz)*Id - rr^T  (packed symmetric)
  float A6[6];
  A6[0] = yy + zz; A6[1] = xx + zz; A6[2] = xx + yy;
  A6[3] = -xy;     A6[4] = -xz;     A6[5] = -yz;

  float lam0, lam1, lam2;
  float V[3][3];  // columns = eigenvectors
  const float q  = (A6[0] + A6[1] + A6[2]) * (1.0f / 3.0f);
  const float b0 = A6[0] - q, b1 = A6[1] - q, b2 = A6[2] - q;
  const float p1 = A6[3] * A6[3] + A6[4] * A6[4] + A6[5] * A6[5];
  const float p2 = b0 * b0 + b1 * b1 + b2 * b2 + 2.0f * p1;

  if (p2 <= 1e-12f * q * q + 1e-30f) {  // ~multiple of identity (incl. zero)
    lam0 = lam1 = lam2 = q;
    V[0][0] = 1.f; V[1][0] = 0.f; V[2][0] = 0.f;
    V[0][1] = 0.f; V[1][1] = 1.f; V[2][1] = 0.f;
    V[0][2] = 0.f; V[1][2] = 0.f; V[2][2] = 1.f;
  } else {
    const float p    = sqrtf(p2 * (1.0f / 6.0f));
    const float invp = 1.0f / p;
    const float B00 = b0 * invp, B11 = b1 * invp, B22 = b2 * invp;
    const float B01 = A6[3] * invp, B02 = A6[4] * invp, B12 = A6[5] * invp;
    const float detB = B00 * (B11 * B22 - B12 * B12)
                     - B01 * (B01 * B22 - B12 * B02)
                     + B02 * (B01 * B12 - B11 * B02);
    const float r   = fminf(fmaxf(0.5f * detB, -1.0f), 1.0f);
    const float phi = acosf(r) * (1.0f / 3.0f);
    lam2 = q + 2.0f * p * cosf(phi);                       // largest
    lam0 = q + 2.0f * p * cosf(phi + 2.0943951023931953f); // smallest
    lam1 = 3.0f * q - lam0 - lam2;

    float v2[3], v0[3], v1[3];
    sym_eigvec(A6, lam2, v2);
    sym_eigvec(A6, lam0, v0);
    // Gram-Schmidt v0 against v2 (robust under near-degeneracy)
    const float d = dot3(v0, v2);
    v0[0] -= d * v2[0]; v0[1] -= d * v2[1]; v0[2] -= d * v2[2];
    float nn = dot3(v0, v0);
    if (nn < 1e-24f) {
      float ax[3] = {1.0f, 0.0f, 0.0f};
      if (fabsf(v2[0]) > 0.9f) { ax[0] = 0.0f; ax[1] = 1.0f; }
      cross3(v2, ax, v0);
      nn = dot3(v0, v0);
    }
    const float inv = rsqrtf(fmaxf(nn, 1e-30f));
    v0[0] *= inv; v0[1] *= inv; v0[2] *= inv;
    cross3(v2, v0, v1);
    V[0][0] = v0[0]; V[1][0] = v0[1]; V[2][0] = v0[2];
    V[0][1] = v1[0]; V[1][1] = v1[1]; V[2][1] = v1[2];
    V[0][2] = v2[0]; V[1][2] = v2[1]; V[2][2] = v2[2];
  }

  const float lam[3] = {lam0, lam1, lam2};
  float maxe = fmaxf(fmaxf(lam0, lam1), lam2);
  maxe = fmaxf(maxe, 1e-8f);
  const bool big = frag_sizes[f] > 1;

  const float t[3] = {tx, ty, tz};
  float omega[3] = {0.f, 0.f, 0.f};
  float P[9] = {0.f, 0.f, 0.f, 0.f, 0.f, 0.f, 0.f, 0.f, 0.f};
#pragma unroll
  for (int i = 0; i < 3; ++i) {
    const float obs = (big && lam[i] > 0.01f * maxe) ? 1.0f : 0.0f;
    const float vi0 = V[0][i], vi1 = V[1][i], vi2 = V[2][i];
    const float te = vi0 * t[0] + vi1 * t[1] + vi2 * t[2];
    const float oe = te / fmaxf(lam[i], 1e-6f) * obs;
    omega[0] += oe * vi0; omega[1] += oe * vi1; omega[2] += oe * vi2;
    P[0] += obs * vi0 * vi0; P[1] += obs * vi0 * vi1; P[2] += obs * vi0 * vi2;
    P[3] += obs * vi1 * vi0; P[4] += obs * vi1 * vi1; P[5] += obs * vi1 * vi2;
    P[6] += obs * vi2 * vi0; P[7] += obs * vi2 * vi1; P[8] += obs * vi2 * vi2;
  }
  w_out[3 * f + 0] = omega[0];
  w_out[3 * f + 1] = omega[1];
  w_out[3 * f + 2] = omega[2];
#pragma unroll
  for (int j = 0; j < 9; ++j) P_out[9 * f + j] = P[j];
}

extern "C" void kernel_launch(void* const* d_in, const int* in_sizes, int n_in,
                              void* d_out, int out_size, void* d_ws, size_t ws_size,
                              hipStream_t stream) {
  const float* f_atom     = (const float*)d_in[0];
  const float* atom_pos   = (const float*)d_in[1];
  const float* T_frag     = (const float*)d_in[2];
  const int*   frag_id    = (const int*)d_in[3];
  const int*   frag_sizes = (const int*)d_in[4];
  const int n_atom = in_sizes[3];
  const int n_frag = in_sizes[4];

  float* acc = (float*)d_ws;              // n_frag * 16 floats (12.8 MB)
  const int acc_n = n_frag * ACC_STRIDE;

  ws_zero<<<(acc_n + 255) / 256, 256, 0, stream>>>(acc, acc_n);
  frag_accumulate<<<(n_atom + 255) / 256, 256, 0, stream>>>(
      f_atom, atom_pos, T_frag, frag_id, acc, n_atom);
  frag_finalize<<<(n_frag + 255) / 256, 256, 0, stream>>>(
      acc, frag_sizes, (float*)d_out, n_frag);

  (void)n_in; (void)out_size; (void)ws_size;
}